// Split_sub_encoder_65704409694829
// MI455X (gfx1250) — compile-verified
//
#include <hip/hip_runtime.h>

typedef float v2f __attribute__((ext_vector_type(2)));
typedef float v8f __attribute__((ext_vector_type(8)));

#define HD 128
#define STATS_BLOCKS 512
#define KP_PITCH 288  // dwords per k-pair row; 288%64==32 -> half-waves use disjoint bank halves

// ---------------- init: zero stat partials + d_out, deg=1 (self loops) ----------------
__global__ void k_init(float* __restrict__ psum2, float* __restrict__ deg,
                       float* __restrict__ out, int N, int outN) {
    int stride = gridDim.x * blockDim.x;
    int i0 = blockIdx.x * blockDim.x + threadIdx.x;
    for (int j = i0; j < 2 * STATS_BLOCKS * HD; j += stride) psum2[j] = 0.0f;
    for (int j = i0; j < N; j += stride) deg[j] = 1.0f;
    for (int j = i0; j < outN; j += stride) out[j] = 0.0f;
}

// ---------------- BN stats: per-column partial sum/sumsq of na*x (deterministic) ------
__global__ void k_stats(const float* __restrict__ x, const float* __restrict__ na,
                        float* __restrict__ psum, float* __restrict__ psumsq, int N) {
    int c = threadIdx.x;                      // blockDim.x == 128
    float s = 0.0f, s2 = 0.0f;
    for (int r = blockIdx.x; r < N; r += gridDim.x) {
        float v = na[r] * x[(size_t)r * HD + c];
        s += v; s2 += v * v;
    }
    psum[blockIdx.x * HD + c]   = s;
    psumsq[blockIdx.x * HD + c] = s2;
}

// ---------------- degree accumulation over edges --------------------------------------
__global__ void k_deg(const int* __restrict__ dst, const float* __restrict__ ea,
                      float* __restrict__ deg, int E) {
    int stride = gridDim.x * blockDim.x;
    for (int e = blockIdx.x * blockDim.x + threadIdx.x; e < E; e += stride)
        atomicAdd(&deg[dst[e]], ea[e]);
}

// ---------------- finalize BN: s=rstd*gamma, t=beta-mu*s, c = t^T W -------------------
__global__ void k_finalize(const float* __restrict__ psum, const float* __restrict__ psumsq,
                           const float* __restrict__ W, const float* __restrict__ gamma,
                           const float* __restrict__ beta, float* __restrict__ sv,
                           float* __restrict__ cv, float invN) {
    __shared__ float sh_t[HD];
    int c = threadIdx.x;                      // 1 block, 128 threads
    float s = 0.0f, s2 = 0.0f;
    for (int b = 0; b < STATS_BLOCKS; ++b) { s += psum[b * HD + c]; s2 += psumsq[b * HD + c]; }
    float mu   = s * invN;
    float var  = s2 * invN - mu * mu;         // biased variance, matches reference
    float rstd = rsqrtf(var + 1e-5f);
    float sc   = rstd * gamma[c];
    float tc   = beta[c] - mu * sc;
    sv[c] = sc; sh_t[c] = tc;
    __syncthreads();
    float acc = 0.0f;
    for (int k = 0; k < HD; ++k) acc += sh_t[k] * W[k * HD + c];
    cv[c] = acc;
}

// ---------------- W2 = S * W (scale rows of W by s[k]) --------------------------------
__global__ void k_w2(const float* __restrict__ W, const float* __restrict__ sv,
                     float* __restrict__ W2) {
    int k = blockIdx.x, n = threadIdx.x;      // 128 x 128
    W2[k * HD + n] = sv[k] * W[k * HD + n];
}

// ---------------- dinv = rsqrt(max(deg,1e-12)) in place -------------------------------
__global__ void k_dinv(float* __restrict__ deg, int N) {
    int stride = gridDim.x * blockDim.x;
    for (int i = blockIdx.x * blockDim.x + threadIdx.x; i < N; i += stride)
        deg[i] = rsqrtf(fmaxf(deg[i], 1e-12f));
}

// ---------------- GEMM: hW[r] = na[r]*(x[r] @ W2) + c, via V_WMMA_F32_16X16X4_F32 -----
// LDS holds W2 K-pair interleaved: (k,n) -> lw[(k>>1)*KP_PITCH + 2n + (k&1)], so each
// lane's B operand (W2[ka][n], W2[ka+1][n]) is a single aligned ds_load_b64.
__global__ void k_gemm(const float* __restrict__ x, const float* __restrict__ na,
                       const float* __restrict__ W2, const float* __restrict__ cv,
                       float* __restrict__ hW, int N, int blockTiles) {
    __shared__ float lw[(HD / 2) * KP_PITCH];
    __shared__ float lc[HD];
    for (int i = threadIdx.x; i < HD * HD; i += blockDim.x) {
        int k = i >> 7, n = i & 127;
        lw[(k >> 1) * KP_PITCH + 2 * n + (k & 1)] = W2[i];
    }
    if (threadIdx.x < HD) lc[threadIdx.x] = cv[threadIdx.x];
    __syncthreads();

    const int wave = threadIdx.x >> 5;        // 8 waves / block
    const int lane = threadIdx.x & 31;
    const int half = lane >> 4;               // lanes 16-31 hold K+2,K+3 halves
    const int l16  = lane & 15;

    float cr[8];
    #pragma unroll
    for (int nt = 0; nt < 8; ++nt) cr[nt] = lc[nt * 16 + l16];

    for (int bt = blockIdx.x; bt < blockTiles; bt += gridDim.x) {
        const int rowbase = (bt * 8 + wave) * 16;
        // clamp A-row reads so EXEC stays all-ones through the WMMA loop
        const int arow = min(rowbase + l16, N - 1);
        const float* xrow = x + (size_t)arow * HD;

        v8f acc[8];
        #pragma unroll
        for (int nt = 0; nt < 8; ++nt) acc[nt] = (v8f){0,0,0,0,0,0,0,0};

        for (int k0 = 0; k0 < HD; k0 += 4) {
            const int ka = k0 + 2 * half;          // A 16x4 f32: lanes0-15 K0/K1, 16-31 K2/K3
            v2f a = *(const v2f*)(xrow + ka);
            const float* brow = &lw[(ka >> 1) * KP_PITCH];
            #pragma unroll
            for (int nt = 0; nt < 8; ++nt) {
                v2f b = *(const v2f*)(brow + 2 * (nt * 16 + l16));  // one ds_load_b64
                acc[nt] = __builtin_amdgcn_wmma_f32_16x16x4_f32(
                    false, a, false, b, (short)0, acc[nt], false, false);
            }
        }

        // D layout: VGPR v -> row v (lanes 0-15) / v+8 (lanes 16-31), col = l16 + nt*16
        #pragma unroll
        for (int v = 0; v < 8; ++v) {
            const int row = rowbase + v + 8 * half;
            if (row < N) {
                const float nav = na[row];
                const size_t ro = (size_t)row * HD;
                #pragma unroll
                for (int nt = 0; nt < 8; ++nt)
                    hW[ro + nt * 16 + l16] = nav * acc[nt][v] + cr[nt];
            }
        }
    }
}

// ---------------- agg init: bias + self-loop contribution -----------------------------
__global__ void k_agginit(const float* __restrict__ hW, const float* __restrict__ dinv,
                          const float* __restrict__ bias, float* __restrict__ agg, int N) {
    int stride = gridDim.x * blockDim.x;
    int total = N * HD;
    for (int i = blockIdx.x * blockDim.x + threadIdx.x; i < total; i += stride) {
        int r = i >> 7, c = i & 127;
        float dv = dinv[r];
        agg[i] = bias[c] + dv * dv * hW[i];
    }
}

// ---------------- edge scatter: wave per edge, float4 gather + 4 f32 atomics ----------
__global__ void k_edges(const int* __restrict__ src, const int* __restrict__ dstv,
                        const float* __restrict__ ea, const float* __restrict__ dinv,
                        const float* __restrict__ hW, float* __restrict__ agg, int E) {
    const int lane = threadIdx.x & 31;
    const int wid  = (blockIdx.x * blockDim.x + threadIdx.x) >> 5;
    const int nw   = (gridDim.x * blockDim.x) >> 5;
    for (int e = wid; e < E; e += nw) {
        const int s = src[e], d = dstv[e];
        const float nrm = dinv[s] * ea[e] * dinv[d];
        const float4 v = ((const float4*)(hW + (size_t)s * HD))[lane];
        float* ad = agg + (size_t)d * HD + lane * 4;
        atomicAdd(ad + 0, nrm * v.x);
        atomicAdd(ad + 1, nrm * v.y);
        atomicAdd(ad + 2, nrm * v.z);
        atomicAdd(ad + 3, nrm * v.w);
    }
}

// ---------------- ReLU + global_add_pool ----------------------------------------------
__global__ void k_pool(const float* __restrict__ agg, const int* __restrict__ batch,
                       float* __restrict__ out, int N) {
    int stride = gridDim.x * blockDim.x;
    int total = N * HD;
    for (int i = blockIdx.x * blockDim.x + threadIdx.x; i < total; i += stride) {
        int r = i >> 7, c = i & 127;
        float v = fmaxf(agg[i], 0.0f);
        atomicAdd(&out[(size_t)batch[r] * HD + c], v);
    }
}

extern "C" void kernel_launch(void* const* d_in, const int* in_sizes, int n_in,
                              void* d_out, int out_size, void* d_ws, size_t ws_size,
                              hipStream_t stream) {
    const float* x     = (const float*)d_in[0];
    const int*   eidx  = (const int*)  d_in[1];
    const int*   batch = (const int*)  d_in[2];
    const float* eatt  = (const float*)d_in[3];
    const float* natt  = (const float*)d_in[4];
    const float* W     = (const float*)d_in[5];
    const float* bias  = (const float*)d_in[6];
    const float* gamma = (const float*)d_in[7];
    const float* beta  = (const float*)d_in[8];
    float* out = (float*)d_out;

    const int N = in_sizes[2];      // batch has N entries
    const int E = in_sizes[3];      // edge_att has E entries
    const int* src = eidx;
    const int* dst = eidx + E;

    // workspace layout (floats)
    float* ws     = (float*)d_ws;
    float* psum   = ws;                               // 512*128
    float* psumsq = psum   + STATS_BLOCKS * HD;       // 512*128
    float* sv     = psumsq + STATS_BLOCKS * HD;       // 128
    float* cv     = sv + HD;                          // 128
    float* W2     = cv + HD;                          // 128*128
    float* deg    = W2 + HD * HD;                     // N (becomes dinv)
    float* hW     = deg + N;                          // N*128
    float* agg    = hW + (size_t)N * HD;              // N*128

    k_init<<<1024, 256, 0, stream>>>(psum, deg, out, N, out_size);
    k_stats<<<STATS_BLOCKS, HD, 0, stream>>>(x, natt, psum, psumsq, N);
    k_deg<<<2048, 256, 0, stream>>>(dst, eatt, deg, E);
    k_finalize<<<1, HD, 0, stream>>>(psum, psumsq, W, gamma, beta, sv, cv, 1.0f / (float)N);
    k_w2<<<HD, HD, 0, stream>>>(W, sv, W2);
    k_dinv<<<512, 256, 0, stream>>>(deg, N);

    const int tiles16 = (N + 15) / 16;
    const int blockTiles = (tiles16 + 7) / 8;
    k_gemm<<<blockTiles, 256, 0, stream>>>(x, natt, W2, cv, hW, N, blockTiles);

    k_agginit<<<2048, 256, 0, stream>>>(hW, deg, bias, agg, N);
    k_edges<<<4096, 256, 0, stream>>>(src, dst, eatt, deg, hW, agg, E);
    k_pool<<<2048, 256, 0, stream>>>(agg, batch, out, N);
}